// vmd_decomp_62483184222299
// MI455X (gfx1250) — compile-verified
//
#include <hip/hip_runtime.h>
#include <math.h>

typedef __attribute__((ext_vector_type(2))) float v2f;
typedef __attribute__((ext_vector_type(8))) float v8f;

#define ALPHA_C 2000.0f
#define TOL_C   1e-7f
#define EPS_C   2.220446049250313e-16f

// ---------------------------------------------------------------------------
// Kernel 0: twiddle table, T[i] = cos(2*pi*i/1024), computed in f64 (exact-ish
// f32 twiddles; avoids the f32 sincos slow-path entirely in the hot kernels).
// sin(2*pi*i/1024) == T[(i+768) & 1023].
// ---------------------------------------------------------------------------
__global__ void vmd_twiddle(float* __restrict__ T) {
  int i = blockIdx.x * blockDim.x + threadIdx.x;   // 0..1023
  double a = (double)i * 6.283185307179586476925286766559e-3 * (1024.0 / 1024.0);
  // angle = 2*pi*i/1024
  a = (double)i * (6.283185307179586476925286766559 / 1024.0);
  T[i] = (float)cos(a);
}

// ---------------------------------------------------------------------------
// Kernel 1: mirror-extend each of the 64 signals into fMT[t][sig] (t-major so
// the DFT B-tiles are row-contiguous).  fM = [f[:256][::-1], f, f[256:][::-1]]
// ---------------------------------------------------------------------------
__global__ void vmd_mirror(const float* __restrict__ x, float* __restrict__ fMT) {
  int idx = blockIdx.x * blockDim.x + threadIdx.x;   // 0..65535
  int t   = idx >> 6;
  int sig = idx & 63;
  int b = sig >> 3, c = sig & 7;
  int s = (t < 256) ? (255 - t) : ((t < 768) ? (t - 256) : (1279 - t));
  fMT[idx] = x[(b * 512 + s) * 8 + c];
}

// ---------------------------------------------------------------------------
// Kernel 2: forward real DFT, upper half-spectrum only (bins m = 0..511):
//   FR[m][sig] =  sum_t fMT[t][sig] * cos(2*pi*m*t/1024)
//   FI[m][sig] = -sum_t fMT[t][sig] * sin(2*pi*m*t/1024)
// GEMM M=512, N=64, K=1024 via V_WMMA_F32_16X16X4_F32, one wave per 16x16 tile.
// Twiddles come from an LDS-resident table indexed by (m*t) mod 1024.
// ---------------------------------------------------------------------------
__global__ void vmd_dft(const float* __restrict__ fMT, const float* __restrict__ Tw,
                        float* __restrict__ FR, float* __restrict__ FI) {
  __shared__ float T[1024];
  for (int i = threadIdx.x; i < 1024; i += 32) T[i] = Tw[i];
  __syncthreads();

  int lane  = threadIdx.x;
  int mtile = blockIdx.x & 31;      // 32 tiles of freq
  int ntile = blockIdx.x >> 5;      // 4  tiles of signal
  int mloc  = lane & 15;
  int kb    = (lane >> 4) << 1;     // 0 for lanes 0-15, 2 for lanes 16-31
  int m     = mtile * 16 + mloc;    // frequency row this lane feeds for A
  int ncol  = ntile * 16 + mloc;    // signal column this lane feeds for B
  v8f accR = {};
  v8f accI = {};
  #pragma unroll 4
  for (int k0 = 0; k0 < 1024; k0 += 4) {
    int t0 = k0 + kb;
    int i0 = (m * t0) & 1023;
    int i1 = (i0 + m) & 1023;       // (m*(t0+1)) mod 1024
    v2f aC = {  T[i0],                T[i1] };
    v2f aS = { -T[(i0 + 768) & 1023], -T[(i1 + 768) & 1023] };
    v2f bv = {  fMT[t0 * 64 + ncol],  fMT[t0 * 64 + 64 + ncol] };
    accR = __builtin_amdgcn_wmma_f32_16x16x4_f32(false, aC, false, bv, (short)0, accR, false, false);
    accI = __builtin_amdgcn_wmma_f32_16x16x4_f32(false, aS, false, bv, (short)0, accI, false, false);
  }
  int rbase = mtile * 16 + ((lane >> 4) << 3);
  int col   = ntile * 16 + (lane & 15);
  #pragma unroll
  for (int j = 0; j < 8; ++j) {
    FR[(rbase + j) * 64 + col] = accR[j];
    FI[(rbase + j) * 64 + col] = accI[j];
  }
}

// ---------------------------------------------------------------------------
// Kernel 3: ADMM iterations.  One block per signal, one lane per frequency
// bin (upper half only; lower half is identically zero since TAU==0).
// Per-iteration cross-lane traffic: 11-value reduction (5 p-sums, 5 wf-sums,
// uDiff) via wave32 shuffles + 16-wave LDS combine.
// ---------------------------------------------------------------------------
__global__ void __launch_bounds__(512)
vmd_iterate(const float* __restrict__ FR, const float* __restrict__ FI,
            float* __restrict__ PR, float* __restrict__ PI) {
  int sig = blockIdx.x;       // 0..63
  int m   = threadIdx.x;      // 0..511  (freq bin; half_freqs[m] = m/1024)
  float freq = (float)m * (1.0f / 1024.0f);
  float fR = FR[m * 64 + sig];
  float fI = FI[m * 64 + sig];
  float uR[5] = {0,0,0,0,0}, uI[5] = {0,0,0,0,0};
  float sR = 0.f, sI = 0.f;
  float omega[5];
  #pragma unroll
  for (int k = 0; k < 5; ++k) omega[k] = 0.1f * (float)k;   // 0.5/K * k
  __shared__ float red[16][12];
  __shared__ float res[12];
  int lane = threadIdx.x & 31;
  int wid  = threadIdx.x >> 5;

  for (int n = 0; n < 499; ++n) {
    sR += uR[4] - uR[0];
    sI += uI[4] - uI[0];
    float part[11];
    float diff = 0.f;
    float prR = 0.f, prI = 0.f;   // previous column's new value
    #pragma unroll
    for (int k = 0; k < 5; ++k) {
      if (k > 0) { sR += prR - uR[k]; sI += prI - uI[k]; }
      float dw  = freq - omega[k];
      float den = 1.0f + ALPHA_C * dw * dw;
      float nR  = (fR - sR) / den;
      float nI  = (fI - sI) / den;
      float pw  = nR * nR + nI * nI;
      part[k]     = pw;
      part[5 + k] = freq * pw;
      float dR = nR - uR[k], dI = nI - uI[k];
      diff += dR * dR + dI * dI;
      uR[k] = nR; uI[k] = nI;
      prR = nR; prI = nI;
    }
    part[10] = diff;
    #pragma unroll
    for (int j = 0; j < 11; ++j) {
      float v = part[j];
      for (int off = 16; off > 0; off >>= 1) v += __shfl_down(v, off, 32);
      part[j] = v;
    }
    if (lane == 0) {
      #pragma unroll
      for (int j = 0; j < 11; ++j) red[wid][j] = part[j];
    }
    __syncthreads();
    if (threadIdx.x < 11) {
      float a = 0.f;
      for (int w = 0; w < 16; ++w) a += red[w][threadIdx.x];
      res[threadIdx.x] = a;
    }
    __syncthreads();
    #pragma unroll
    for (int k = 0; k < 5; ++k) omega[k] = res[5 + k] / res[k];
    float ud = fabsf(EPS_C + res[10] * (1.0f / 1024.0f));
    if (ud <= TOL_C) break;     // uniform across block (res is shared)
  }
  #pragma unroll
  for (int k = 0; k < 5; ++k) {
    PR[m * 320 + sig * 5 + k] = uR[k];
    PI[m * 320 + sig * 5 + k] = uI[k];
  }
}

// ---------------------------------------------------------------------------
// Kernel 4: inverse transform (the real IFFT collapses algebraically):
//   Ut[s][c] = (1/1024) * [ ReP[0] + (-1)^s ReP[511]
//              + 2*sum_{j=1..511}( ReP[j] cos(th) - ImP[j] sin(th) ) ]
// with th = 2*pi*j*(s+256)/1024.  GEMM M=512, N=320, K=512 via f32 WMMA.
// ---------------------------------------------------------------------------
__global__ void vmd_idft(const float* __restrict__ PR, const float* __restrict__ PI,
                         const float* __restrict__ Tw, float* __restrict__ Ut) {
  __shared__ float T[1024];
  for (int i = threadIdx.x; i < 1024; i += 32) T[i] = Tw[i];
  __syncthreads();

  int lane  = threadIdx.x;
  int mtile = blockIdx.x & 31;      // 32 tiles of s
  int ntile = blockIdx.x >> 5;      // 20 tiles of (sig,mode) columns
  int sloc  = lane & 15;
  int kb    = (lane >> 4) << 1;
  int srow  = mtile * 16 + sloc;    // s = 0..511, actual time tt = s+256
  int tt    = srow + 256;
  float sgn = (srow & 1) ? -1.0f : 1.0f;   // (-1)^tt == (-1)^s (256 even)
  int ncol  = ntile * 16 + sloc;    // 0..319
  v8f acc = {};
  #pragma unroll 4
  for (int k0 = 0; k0 < 512; k0 += 4) {
    int j0 = k0 + kb;
    int i0 = (j0 * tt) & 1023;
    int i1 = (i0 + tt) & 1023;      // ((j0+1)*tt) mod 1024
    float aC0 =  2.0f * T[i0],                aC1 =  2.0f * T[i1];
    float aS0 = -2.0f * T[(i0 + 768) & 1023], aS1 = -2.0f * T[(i1 + 768) & 1023];
    if (j0 == 0)       { aC0 = 1.0f; aS0 = 0.0f; }  // coefficient of Re pos[0]
    if (j0 + 1 == 511) { aC1 += sgn; }              // extra (-1)^t Re pos[511]
    v2f aC = {aC0, aC1};
    v2f aS = {aS0, aS1};
    v2f bR = { PR[j0 * 320 + ncol], PR[j0 * 320 + 320 + ncol] };
    v2f bI = { PI[j0 * 320 + ncol], PI[j0 * 320 + 320 + ncol] };
    acc = __builtin_amdgcn_wmma_f32_16x16x4_f32(false, aC, false, bR, (short)0, acc, false, false);
    acc = __builtin_amdgcn_wmma_f32_16x16x4_f32(false, aS, false, bI, (short)0, acc, false, false);
  }
  int rbase = mtile * 16 + ((lane >> 4) << 3);
  int col   = ntile * 16 + (lane & 15);
  #pragma unroll
  for (int j = 0; j < 8; ++j)
    Ut[(rbase + j) * 320 + col] = acc[j] * (1.0f / 1024.0f);
}

// ---------------------------------------------------------------------------
// Kernel 5: assemble outputs.  d_out = [seasonal (8*512*1) | trend (8*512*8)].
// trend[b][s][c] = mode0 of signal b*8+c.  seasonal reproduces the reference's
// flat reshape of modes[1:] (4,8,512,8) -> (8,512,32) then mean over last axis.
// ---------------------------------------------------------------------------
__global__ void vmd_assemble(const float* __restrict__ Ut, float* __restrict__ out) {
  int idx = blockIdx.x * blockDim.x + threadIdx.x;
  if (idx >= 36864) return;
  if (idx < 4096) {
    int b1 = idx >> 9;
    int s1 = idx & 511;
    int base = b1 * 16384 + s1 * 32;
    float acc = 0.f;
    for (int j = 0; j < 32; ++j) {
      int p  = base + j;              // flat index into (k-1, b, s, c)
      int k1 = p >> 15;               // mode-1 (0..3)
      int b  = (p >> 12) & 7;
      int s  = (p >> 3) & 511;
      int c  = p & 7;
      acc += Ut[s * 320 + (b * 8 + c) * 5 + (k1 + 1)];
    }
    out[idx] = acc * (1.0f / 32.0f);
  } else {
    int f = idx - 4096;
    int b = f >> 12;
    int s = (f >> 3) & 511;
    int c = f & 7;
    out[idx] = Ut[s * 320 + (b * 8 + c) * 5];   // mode 0 = trend
  }
}

// ---------------------------------------------------------------------------
extern "C" void kernel_launch(void* const* d_in, const int* in_sizes, int n_in,
                              void* d_out, int out_size, void* d_ws, size_t ws_size,
                              hipStream_t stream) {
  (void)in_sizes; (void)n_in; (void)out_size; (void)ws_size;
  const float* x = (const float*)d_in[0];
  float* ws  = (float*)d_ws;
  float* fMT = ws;                         // 1024*64   = 65536
  float* FR  = ws + 65536;                 // 512*64    = 32768
  float* FI  = ws + 98304;                 // 512*64    = 32768
  float* PR  = ws + 131072;                // 512*320   = 163840
  float* PI  = ws + 294912;                // 512*320   = 163840
  float* Ut  = ws + 458752;                // 512*320   = 163840
  float* Tw  = ws + 622592;                // 1024      (total ~2.5 MB)
  float* out = (float*)d_out;

  vmd_twiddle <<<  4, 256, 0, stream>>>(Tw);
  vmd_mirror  <<<256, 256, 0, stream>>>(x, fMT);
  vmd_dft     <<<128,  32, 0, stream>>>(fMT, Tw, FR, FI);
  vmd_iterate <<< 64, 512, 0, stream>>>(FR, FI, PR, PI);
  vmd_idft    <<<640,  32, 0, stream>>>(PR, PI, Tw, Ut);
  vmd_assemble<<<144, 256, 0, stream>>>(Ut, out);
}